// GMEdgeConv_5385888989487
// MI455X (gfx1250) — compile-verified
//
#include <hip/hip_runtime.h>

typedef __attribute__((ext_vector_type(2))) float v2f;
typedef __attribute__((ext_vector_type(8))) float v8f;

#define APAD 132   // 16x128 A tile, padded row stride (132 % 64 == 4 -> conflict-free)
#define HPAD 68    // 16x64 h1 tile, padded row stride

// D = A(16x4,f32) * B(4x16,f32) + C(16x16,f32) on the CDNA5 matrix pipe.
__device__ __forceinline__ v8f wmma_f32_k4(v2f a, v2f b, v8f c) {
  return __builtin_amdgcn_wmma_f32_16x16x4_f32(
      /*neg_a=*/false, a, /*neg_b=*/false, b,
      /*c_mod=*/(short)0, c, /*reuse_a=*/false, /*reuse_b=*/false);
}

__global__ void gme_zero_kernel(unsigned int* __restrict__ p, long long n) {
  long long i = (long long)blockIdx.x * blockDim.x + threadIdx.x;
  long long stride = (long long)gridDim.x * blockDim.x;
  for (; i < n; i += stride) p[i] = 0u;
}

// Persistent block = 4 waves cooperating on one 16-edge tile per iteration.
// Wave w owns output columns [16w, 16w+16). Weight fragments live in VGPRs.
// blockIdx.y selects branch (0 = topo, 1 = geo).
__global__ __launch_bounds__(128) void gme_edge_kernel(
    const float* __restrict__ x,
    const int* __restrict__ ei_topo, const int* __restrict__ ei_geo,
    const float* __restrict__ Wt1, const float* __restrict__ bt1,
    const float* __restrict__ Wt2, const float* __restrict__ bt2,
    const float* __restrict__ Wg1, const float* __restrict__ bg1,
    const float* __restrict__ Wg2, const float* __restrict__ bg2,
    unsigned int* __restrict__ pooled_base,
    int E, int N, int tiles)
{
  __shared__ float As[16 * APAD];  // [16 edges][xi | xj (-> xj-xi)], padded
  __shared__ float Hs[16 * HPAD];  // layer-1 activations, padded
  __shared__ int   Is[16];         // target node per edge row

  const int branch = blockIdx.y;
  const int*   ei = branch ? ei_geo : ei_topo;
  const float* W1 = branch ? Wg1 : Wt1;
  const float* b1 = branch ? bg1 : bt1;
  const float* W2 = branch ? Wg2 : Wt2;
  const float* b2 = branch ? bg2 : bt2;
  unsigned int* pooled = pooled_base + (size_t)branch * (size_t)N * 64;

  const int tid  = threadIdx.x;
  const int wave = tid >> 5;        // n-tile owned by this wave
  const int lane = tid & 31;
  const int m    = lane & 15;
  const int half = lane >> 4;
  const int Nc   = wave * 16 + m;   // output column owned by this lane

  // ---- Preload weight fragments into VGPRs (amortized over all tiles) ----
  v2f w1f[32], w2f[16];
  #pragma unroll
  for (int s = 0; s < 32; ++s) {
    const int c = 4 * s + 2 * half;
    w1f[s].x = W1[c * 64 + Nc];
    w1f[s].y = W1[(c + 1) * 64 + Nc];
  }
  #pragma unroll
  for (int s = 0; s < 16; ++s) {
    const int c = 4 * s + 2 * half;
    w2f[s].x = W2[c * 64 + Nc];
    w2f[s].y = W2[(c + 1) * 64 + Nc];
  }
  const float bias1 = b1[Nc];
  const float bias2 = b2[Nc];

  // Staging role: thread -> (row-half rh, quarter q); 32 row-halves x 4 quarters.
  const int rh = tid >> 2;          // 0..31
  const int q  = tid & 3;           // 16-float quarter of a 64-float row
  const int sm = rh & 15;           // edge row staged
  const int sh = rh >> 4;           // 0: xi, 1: xj

  for (int tile = blockIdx.x; tile < tiles; tile += gridDim.x) {
    int e = tile * 16 + sm;
    if (e >= E) e = E - 1;                 // idempotent under max
    const int node = ei[sh * E + e];
    if ((q | sh) == 0) Is[sm] = node;
    {  // prefetch next tile's edge indices (gfx1250 global_prefetch_b8)
      int en = e + 16 * gridDim.x;
      if (en < E) __builtin_prefetch(&ei[sh * E + en], 0, 0);
    }
    {  // gather node row quarter into LDS (L2-resident), float4-vectorized
      const float4* src = (const float4*)(x + (size_t)node * 64) + q * 4;
      float4* dst = (float4*)(&As[sm * APAD + sh * 64]) + q * 4;
      #pragma unroll
      for (int t = 0; t < 4; ++t) dst[t] = src[t];
    }
    __syncthreads();

    // cols 64..127 := xj - xi  (16*64 elems, 8 per thread)
    #pragma unroll
    for (int t = 0; t < 8; ++t) {
      const int pos = tid + t * 128;
      const int r = pos >> 6, c = pos & 63;
      As[r * APAD + 64 + c] -= As[r * APAD + c];
    }
    __syncthreads();

    // ---- Layer 1: this wave's 16 output cols, K = 128, 32 WMMAs ----
    v8f acc;
    #pragma unroll
    for (int r = 0; r < 8; ++r) acc[r] = bias1;
    #pragma unroll
    for (int s = 0; s < 32; ++s) {
      const int c = 4 * s + 2 * half;
      v2f a;
      a.x = As[m * APAD + c];
      a.y = As[m * APAD + c + 1];
      acc = wmma_f32_k4(a, w1f[s], acc);
    }
    #pragma unroll
    for (int r = 0; r < 8; ++r) {
      const float v = acc[r] > 0.f ? acc[r] : 0.f;   // ReLU
      Hs[(r + 8 * half) * HPAD + Nc] = v;
    }
    __syncthreads();

    // ---- Layer 2: K = 64, 16 WMMAs, then atomic segment-max scatter ----
    v8f acc2;
    #pragma unroll
    for (int r = 0; r < 8; ++r) acc2[r] = bias2;
    #pragma unroll
    for (int s = 0; s < 16; ++s) {
      const int c = 4 * s + 2 * half;
      v2f a;
      a.x = Hs[m * HPAD + c];
      a.y = Hs[m * HPAD + c + 1];
      acc2 = wmma_f32_k4(a, w2f[s], acc2);
    }
    #pragma unroll
    for (int r = 0; r < 8; ++r) {
      const float v = acc2[r] > 0.f ? acc2[r] : 0.f;  // ReLU => v >= 0
      const int ni = Is[r + 8 * half];
      // nonneg float bits order like uint; pooled init 0 == 0.0f matches clamp
      atomicMax(&pooled[(size_t)ni * 64 + Nc], __float_as_uint(v));
    }
    __syncthreads();   // before next iteration overwrites As/Hs/Is
  }
}

// Persistent block = 4 waves on one 16-node tile: relu([topo, geo] Wf + bf).
__global__ __launch_bounds__(128) void gme_final_kernel(
    const float* __restrict__ pt, const float* __restrict__ pg,
    const float* __restrict__ Wf, const float* __restrict__ bf,
    float* __restrict__ out, int N, int tiles)
{
  __shared__ float As[16 * APAD];

  const int tid  = threadIdx.x;
  const int wave = tid >> 5;
  const int lane = tid & 31;
  const int m    = lane & 15;
  const int half = lane >> 4;
  const int Nc   = wave * 16 + m;

  v2f wff[32];
  #pragma unroll
  for (int s = 0; s < 32; ++s) {
    const int c = 4 * s + 2 * half;
    wff[s].x = Wf[c * 64 + Nc];
    wff[s].y = Wf[(c + 1) * 64 + Nc];
  }
  const float bias = bf[Nc];

  const int rh = tid >> 2;
  const int q  = tid & 3;
  const int sm = rh & 15;
  const int sh = rh >> 4;

  for (int tile = blockIdx.x; tile < tiles; tile += gridDim.x) {
    int row = tile * 16 + sm;
    if (row >= N) row = N - 1;
    {
      const float* srcp = (sh ? pg : pt) + (size_t)row * 64;
      const float4* src = (const float4*)srcp + q * 4;
      float4* dst = (float4*)(&As[sm * APAD + sh * 64]) + q * 4;
      #pragma unroll
      for (int t = 0; t < 4; ++t) dst[t] = src[t];
    }
    __syncthreads();

    v8f acc;
    #pragma unroll
    for (int r = 0; r < 8; ++r) acc[r] = bias;
    #pragma unroll
    for (int s = 0; s < 32; ++s) {
      const int c = 4 * s + 2 * half;
      v2f a;
      a.x = As[m * APAD + c];
      a.y = As[m * APAD + c + 1];
      acc = wmma_f32_k4(a, wff[s], acc);
    }
    #pragma unroll
    for (int r = 0; r < 8; ++r) {
      const float v = acc[r] > 0.f ? acc[r] : 0.f;
      const int orow = tile * 16 + r + 8 * half;
      if (orow < N) out[(size_t)orow * 64 + Nc] = v;
    }
    __syncthreads();
  }
}

extern "C" void kernel_launch(void* const* d_in, const int* in_sizes, int n_in,
                              void* d_out, int out_size, void* d_ws, size_t ws_size,
                              hipStream_t stream) {
  const float* x   = (const float*)d_in[0];
  const int*   eit = (const int*)d_in[1];
  const int*   eig = (const int*)d_in[2];
  const float* Wt1 = (const float*)d_in[3];
  const float* bt1 = (const float*)d_in[4];
  const float* Wt2 = (const float*)d_in[5];
  const float* bt2 = (const float*)d_in[6];
  const float* Wg1 = (const float*)d_in[7];
  const float* bg1 = (const float*)d_in[8];
  const float* Wg2 = (const float*)d_in[9];
  const float* bg2 = (const float*)d_in[10];
  const float* Wf  = (const float*)d_in[11];
  const float* bf  = (const float*)d_in[12];

  const int N = in_sizes[0] / 64;   // nodes
  const int E = in_sizes[1] / 2;    // edges per branch

  unsigned int* pooled = (unsigned int*)d_ws;          // [2][N][64]
  const long long pooled_elems = 2LL * N * 64;

  {  // pooled = 0 every call (ws is not re-poisoned between replays)
    long long b = (pooled_elems + 255) / 256;
    int blocks = (int)(b > 50000 ? 50000 : b);
    gme_zero_kernel<<<blocks, 256, 0, stream>>>(pooled, pooled_elems);
  }
  {  // both branches: edge MLP + segment-max (persistent blocks)
    int tiles = (E + 15) / 16;
    int bx = tiles < 8192 ? tiles : 8192;
    dim3 grid(bx, 2);
    gme_edge_kernel<<<grid, 128, 0, stream>>>(
        x, eit, eig, Wt1, bt1, Wt2, bt2, Wg1, bg1, Wg2, bg2,
        pooled, E, N, tiles);
  }
  {  // final fused GEMM + ReLU (persistent blocks)
    int tiles = (N + 15) / 16;
    int bx = tiles < 4096 ? tiles : 4096;
    gme_final_kernel<<<bx, 128, 0, stream>>>(
        (const float*)pooled, (const float*)pooled + (size_t)N * 64,
        Wf, bf, (float*)d_out, N, tiles);
  }
}